// SpectralConv2d_7533372637282
// MI455X (gfx1250) — compile-verified
//
#include <hip/hip_runtime.h>

// ---------------------------------------------------------------------------
// SpectralConv2d on gfx1250: truncated-DFT-as-GEMM using V_WMMA_F32_16X16X4_F32
// x: [16,64,256,256] f32. Only kx in {0..31, 224..255} (stored as j=0..63) and
// ky in {0..31} spectral modes are live, so every FFT stage is a small GEMM
// against a twiddle table, executed on the fp32 WMMA path.
// ---------------------------------------------------------------------------

typedef __attribute__((ext_vector_type(2))) float v2f;
typedef __attribute__((ext_vector_type(8))) float v8f;

__device__ __forceinline__ v8f wmma4(v2f a, v2f b, v8f c) {
  // D = A(16x4) * B(4x16) + C(16x16), fp32
  return __builtin_amdgcn_wmma_f32_16x16x4_f32(false, a, false, b, (short)0, c,
                                               false, false);
}

#define NB 16
#define NC 64
#define NS 256
#define NKY 32
#define NKX 64  // 32 low + 32 high kx modes (storage index j)
#define TWO_PI 6.283185307179586f

// ---- workspace layout (float offsets; first 64 doubles = GN stat accum) ----
constexpr size_t NF_BROW = 256 * 32;
constexpr size_t NF_CCOL = 64 * 256;
constexpr size_t NF_FCOL = 256 * 64;
constexpr size_t NF_GROW = 32 * 256;
constexpr size_t NF_Y1 = (size_t)NB * NC * NS * NKY;   // 8388608
constexpr size_t NF_Y2 = (size_t)NB * NC * NKX * NKY;  // 2097152

constexpr size_t OFF_STATS1 = 128;  // 64 doubles precede (gsum1[32], gsum2[32])
constexpr size_t OFF_STATS2 = OFF_STATS1 + 32;
constexpr size_t OFF_TBROW_RE = 192;
constexpr size_t OFF_TBROW_IM = OFF_TBROW_RE + NF_BROW;
constexpr size_t OFF_TCCOL_RE = OFF_TBROW_IM + NF_BROW;
constexpr size_t OFF_TCCOL_IM = OFF_TCCOL_RE + NF_CCOL;
constexpr size_t OFF_TFCOL_RE = OFF_TCCOL_IM + NF_CCOL;
constexpr size_t OFF_TFCOL_IM = OFF_TFCOL_RE + NF_FCOL;
constexpr size_t OFF_TGROW_RE = OFF_TFCOL_IM + NF_FCOL;
constexpr size_t OFF_TGROW_IM = OFF_TGROW_RE + NF_GROW;
constexpr size_t OFF_Y1RE = OFF_TGROW_IM + NF_GROW;
constexpr size_t OFF_Y1IM = OFF_Y1RE + NF_Y1;
constexpr size_t OFF_Y2RE = OFF_Y1IM + NF_Y1;
constexpr size_t OFF_Y2IM = OFF_Y2RE + NF_Y2;
constexpr size_t OFF_FEAT = OFF_Y2IM + NF_Y2;
constexpr size_t OFF_ORE = OFF_FEAT + NF_Y2;
constexpr size_t OFF_OIM = OFF_ORE + NF_Y2;
// Z aliases Y1 (Y1 dead after column DFT, Z born at inverse column DFT)
constexpr size_t OFF_ZRE = OFF_Y1RE;
constexpr size_t OFF_ZIM = OFF_Y1IM;

// ---------------------------------------------------------------------------
// Twiddle tables (rebuilt every call: deterministic, no cached state).
// ---------------------------------------------------------------------------
__global__ void k_tables(float* ws) {
  int t = blockIdx.x * 256 + threadIdx.x;  // 16384 threads
  if (t < 8192) {
    // forward row rfft: B[y][ky] = e^{-2πi ky y/256}
    int y = t >> 5, ky = t & 31;
    float th = (float)((y * ky) & 255) * (TWO_PI / 256.0f);
    ws[OFF_TBROW_RE + t] = cosf(th);
    ws[OFF_TBROW_IM + t] = -sinf(th);
    // inverse row irfft: G[ky][yy] = c_ky * e^{+2πi ky yy/256} / 65536, Re-proj
    int kyg = t >> 8, yy = t & 255;
    float c = ((kyg == 0) ? 1.0f : 2.0f) / 65536.0f;
    float thg = (float)((kyg * yy) & 255) * (TWO_PI / 256.0f);
    ws[OFF_TGROW_RE + t] = c * cosf(thg);
    ws[OFF_TGROW_IM + t] = -c * sinf(thg);
  }
  if (t < 16384) {
    // forward column DFT A: C[j][x] = e^{-2πi kx(j) x/256}
    int j = t >> 8, xx = t & 255;
    int kx = (j < 32) ? j : (192 + j);
    float th = (float)((kx * xx) & 255) * (TWO_PI / 256.0f);
    ws[OFF_TCCOL_RE + t] = cosf(th);
    ws[OFF_TCCOL_IM + t] = -sinf(th);
    // inverse column DFT A: F[x][j] = e^{+2πi kx(j) x/256}
    int x2 = t >> 6, j2 = t & 63;
    int kx2 = (j2 < 32) ? j2 : (192 + j2);
    float th2 = (float)((kx2 * x2) & 255) * (TWO_PI / 256.0f);
    ws[OFF_TFCOL_RE + t] = cosf(th2);
    ws[OFF_TFCOL_IM + t] = sinf(th2);
  }
}

// ---------------------------------------------------------------------------
// GroupNorm statistics (double accumulation, per-batch)
// ---------------------------------------------------------------------------
__device__ __forceinline__ void block_reduce_atomic(double s, double s2,
                                                    double* d0, double* d1) {
  __shared__ double sh0[256];
  __shared__ double sh1[256];
  int t = threadIdx.x;
  sh0[t] = s;
  sh1[t] = s2;
  __syncthreads();
  for (int off = 128; off > 0; off >>= 1) {
    if (t < off) {
      sh0[t] += sh0[t + off];
      sh1[t] += sh1[t + off];
    }
    __syncthreads();
  }
  if (t == 0) {
    atomicAdd(d0, sh0[0]);
    atomicAdd(d1, sh1[0]);
  }
}

__global__ void k_stats1(const float* __restrict__ x, double* gsum) {
  int b = blockIdx.y;
  const float* xb = x + (size_t)b * 4194304;
  double s = 0.0, s2 = 0.0;
  for (size_t i = (size_t)blockIdx.x * blockDim.x + threadIdx.x; i < 4194304;
       i += (size_t)gridDim.x * blockDim.x) {
    float v = xb[i];
    s += (double)v;
    s2 += (double)v * (double)v;
  }
  block_reduce_atomic(s, s2, &gsum[b * 2], &gsum[b * 2 + 1]);
}

__global__ void k_finalize(const double* gsum, float* stats, double inv_count) {
  int b = threadIdx.x;
  if (b < NB) {
    double m = gsum[b * 2] * inv_count;
    double v = gsum[b * 2 + 1] * inv_count - m * m;
    stats[b * 2] = (float)m;
    stats[b * 2 + 1] = (float)(1.0 / sqrt(v + 1e-5));
  }
}

// ---------------------------------------------------------------------------
// Stage B: fused GN1-normalize + row rfft (keep ky=0..31).
// GEMM: D[row][ky] = sum_y xn[row][y] * T[y][ky].  One wave = 16 rows x 32 ky.
// ---------------------------------------------------------------------------
__global__ void k_rowdft(const float* __restrict__ x,
                         const float* __restrict__ stats,
                         const float* __restrict__ gw,
                         const float* __restrict__ gb,
                         const float* __restrict__ tRe,
                         const float* __restrict__ tIm, float* __restrict__ Y1re,
                         float* __restrict__ Y1im) {
  int wave = blockIdx.x * 8 + (threadIdx.x >> 5);  // 16384 tiles
  int lane = threadIdx.x & 31;
  int l16 = lane & 15, half = lane >> 4;
  int row0 = wave * 16;
  int myrow = row0 + l16;       // (b*64+c)*256 + xrow
  int b = myrow >> 14;
  int c = (myrow >> 8) & 63;
  float mu = stats[b * 2], istd = stats[b * 2 + 1];
  float sc = istd * gw[c];
  float bs = gb[c] - mu * sc;
  const float* xr = x + (size_t)myrow * NS;
  v8f aR0 = {}, aI0 = {}, aR1 = {}, aI1 = {};
  for (int k0 = 0; k0 < NS; k0 += 4) {
    const float* ap = xr + k0 + 2 * half;
    __builtin_prefetch(xr + k0 + 64, 0, 0);
    v2f a;
    a.x = ap[0] * sc + bs;
    a.y = ap[1] * sc + bs;
    int kr = k0 + 2 * half;
    v2f br0, bi0, br1, bi1;
    br0.x = tRe[kr * 32 + l16];
    br0.y = tRe[(kr + 1) * 32 + l16];
    bi0.x = tIm[kr * 32 + l16];
    bi0.y = tIm[(kr + 1) * 32 + l16];
    br1.x = tRe[kr * 32 + 16 + l16];
    br1.y = tRe[(kr + 1) * 32 + 16 + l16];
    bi1.x = tIm[kr * 32 + 16 + l16];
    bi1.y = tIm[(kr + 1) * 32 + 16 + l16];
    aR0 = wmma4(a, br0, aR0);
    aI0 = wmma4(a, bi0, aI0);
    aR1 = wmma4(a, br1, aR1);
    aI1 = wmma4(a, bi1, aI1);
  }
  for (int r = 0; r < 8; ++r) {
    int rr = row0 + r + 8 * half;
    size_t o = (size_t)rr * NKY + l16;
    Y1re[o] = aR0[r];
    Y1im[o] = aI0[r];
    Y1re[o + 16] = aR1[r];
    Y1im[o + 16] = aI1[r];
  }
}

// ---------------------------------------------------------------------------
// Stage C: column DFT. Per (b,c): D[j][ky] = sum_x T[j][x] * Y1[x][ky], complex.
// One block = one (b,c); 8 waves = 4 M-tiles x 2 N-tiles.
// ---------------------------------------------------------------------------
__global__ void k_coldft(const float* __restrict__ Y1re,
                         const float* __restrict__ Y1im,
                         const float* __restrict__ Are,
                         const float* __restrict__ Aim, float* __restrict__ Y2re,
                         float* __restrict__ Y2im) {
  int bc = blockIdx.x;
  int w = threadIdx.x >> 5;
  int mt = w >> 1, nt = w & 1;
  int lane = threadIdx.x & 31, l16 = lane & 15, half = lane >> 4;
  const float* Br = Y1re + (size_t)bc * (NS * NKY);
  const float* Bi = Y1im + (size_t)bc * (NS * NKY);
  int arow = mt * 16 + l16;   // j (kx storage index)
  int ncol = nt * 16 + l16;   // ky
  v8f aR = {}, aI = {};
  for (int k0 = 0; k0 < NS; k0 += 4) {
    int kk = k0 + 2 * half;
    v2f ar = *reinterpret_cast<const v2f*>(Are + arow * NS + kk);
    v2f ai = *reinterpret_cast<const v2f*>(Aim + arow * NS + kk);
    v2f ain = {-ai.x, -ai.y};
    v2f br, bi;
    br.x = Br[(size_t)kk * NKY + ncol];
    br.y = Br[(size_t)(kk + 1) * NKY + ncol];
    bi.x = Bi[(size_t)kk * NKY + ncol];
    bi.y = Bi[(size_t)(kk + 1) * NKY + ncol];
    aR = wmma4(ar, br, aR);
    aR = wmma4(ain, bi, aR);  // re = Ar*Br - Ai*Bi
    aI = wmma4(ar, bi, aI);
    aI = wmma4(ai, br, aI);   // im = Ar*Bi + Ai*Br
  }
  for (int r = 0; r < 8; ++r) {
    int j = mt * 16 + r + 8 * half;
    size_t o = (size_t)bc * (NKX * NKY) + (size_t)j * NKY + ncol;
    Y2re[o] = aR[r];
    Y2im[o] = aI[r];
  }
}

// ---------------------------------------------------------------------------
// Stage D1: feature MLP (15 -> 64 -> gelu -> 64), per (b, j, ky) position.
// ---------------------------------------------------------------------------
__global__ void k_feature(const float* __restrict__ ReN,
                          const float* __restrict__ w1,
                          const float* __restrict__ b1,
                          const float* __restrict__ w2,
                          const float* __restrict__ b2,
                          float* __restrict__ feat) {
  int t = blockIdx.x * 256 + threadIdx.x;  // 16*64*32 = 32768
  if (t >= NB * NKX * NKY) return;
  int ky = t & 31, j = (t >> 5) & 63, b = t >> 11;
  int kx = (j < 32) ? j : (192 + j);
  float k1 = (kx < 128) ? (float)kx : (float)(256 - kx);
  float k2 = (float)ky;
  const float pw[5] = {-1.0f, -0.6f, -0.2f, 0.2f, 0.6f};
  float fin[15];
  for (int f = 0; f < 5; ++f) {
    fin[2 * f + 0] = (k1 == 0.0f) ? 0.0f : powf(k1, pw[f]);
    fin[2 * f + 1] = (k2 == 0.0f) ? 0.0f : powf(k2, pw[f]);
  }
  float re = ReN[b];
  for (int f = 0; f < 5; ++f) fin[10 + f] = powf(re, pw[f]);
  float h[64];
  for (int o = 0; o < 64; ++o) {
    float s = b1[o];
    for (int i = 0; i < 15; ++i) s += w1[o * 15 + i] * fin[i];
    h[o] = 0.5f * s * (1.0f + erff(s * 0.70710678118654752f));  // exact gelu
  }
  for (int o = 0; o < 64; ++o) {
    float s = b2[o];
    for (int i = 0; i < 64; ++i) s += w2[o * 64 + i] * h[i];
    feat[(((size_t)b * NC + o) * NKX + j) * NKY + ky] = s;
  }
}

// Stage D2: elementwise spectral scaling x_ft *= feature (same layout).
__global__ void k_mulfeat(float* __restrict__ ws) {
  size_t t = (size_t)blockIdx.x * 256 + threadIdx.x;  // 2097152
  float f = ws[OFF_FEAT + t];
  ws[OFF_Y2RE + t] *= f;
  ws[OFF_Y2IM + t] *= f;
}

// ---------------------------------------------------------------------------
// Stage E: per-(j,ky) complex channel mix: O[b][o] = sum_i X[b][i] * W[i][o].
// One wave: M=16 (batch) x N=16 (o-tile), K=64 (i). 4 N-tiles per position.
// ---------------------------------------------------------------------------
__global__ void k_specmul(const float* __restrict__ Xre,
                          const float* __restrict__ Xim,
                          const float2* __restrict__ w3,
                          const float2* __restrict__ w4,
                          float* __restrict__ Ore, float* __restrict__ Oim) {
  int gw = blockIdx.x * 8 + (threadIdx.x >> 5);  // 8192 waves
  int nt = gw & 3;
  int pos = gw >> 2;       // j*32 + ky
  int ky = pos & 31, j = pos >> 5;
  int lane = threadIdx.x & 31, l16 = lane & 15, half = lane >> 4;
  const float2* W = (j < 32) ? (w3 + ((size_t)j * NKY + ky))
                             : (w4 + ((size_t)(j - 32) * NKY + ky));
  int ocol = nt * 16 + l16;
  v8f aR = {}, aI = {};
  for (int k0 = 0; k0 < NC; k0 += 4) {
    int kk = k0 + 2 * half;
    size_t abase = (((size_t)l16 * NC + kk) * NKX + j) * NKY + ky;
    v2f ar, ai;
    ar.x = Xre[abase];
    ar.y = Xre[abase + (size_t)NKX * NKY];
    ai.x = Xim[abase];
    ai.y = Xim[abase + (size_t)NKX * NKY];
    v2f ain = {-ai.x, -ai.y};
    float2 w0 = W[((size_t)kk * NC + ocol) * (NKX / 2 * NKY)];        // (i,o) stride = 32*32
    float2 w1 = W[((size_t)(kk + 1) * NC + ocol) * (NKX / 2 * NKY)];
    v2f br = {w0.x, w1.x};
    v2f bi = {w0.y, w1.y};
    aR = wmma4(ar, br, aR);
    aR = wmma4(ain, bi, aR);
    aI = wmma4(ar, bi, aI);
    aI = wmma4(ai, br, aI);
  }
  for (int r = 0; r < 8; ++r) {
    int bb = r + 8 * half;
    size_t o = (((size_t)bb * NC + ocol) * NKX + j) * NKY + ky;
    Ore[o] = aR[r];
    Oim[o] = aI[r];
  }
}

// ---------------------------------------------------------------------------
// Stage F: inverse column DFT. Per (b,o): Z[x][ky] = sum_j F[x][j]*O[j][ky].
// 32 waves per (b,o): 16 M-tiles x 2 N-tiles.
// ---------------------------------------------------------------------------
__global__ void k_icoldft(const float* __restrict__ Ore,
                          const float* __restrict__ Oim,
                          const float* __restrict__ Are,
                          const float* __restrict__ Aim,
                          float* __restrict__ Zre, float* __restrict__ Zim) {
  int gw = blockIdx.x * 8 + (threadIdx.x >> 5);  // 32768 waves
  int bo = gw >> 5;
  int tt = gw & 31;
  int mt = tt >> 1, nt = tt & 1;
  int lane = threadIdx.x & 31, l16 = lane & 15, half = lane >> 4;
  const float* Br = Ore + (size_t)bo * (NKX * NKY);
  const float* Bi = Oim + (size_t)bo * (NKX * NKY);
  int arow = mt * 16 + l16;  // x
  int ncol = nt * 16 + l16;  // ky
  v8f aR = {}, aI = {};
  for (int k0 = 0; k0 < NKX; k0 += 4) {
    int kk = k0 + 2 * half;
    v2f ar = *reinterpret_cast<const v2f*>(Are + arow * NKX + kk);
    v2f ai = *reinterpret_cast<const v2f*>(Aim + arow * NKX + kk);
    v2f ain = {-ai.x, -ai.y};
    v2f br, bi;
    br.x = Br[(size_t)kk * NKY + ncol];
    br.y = Br[(size_t)(kk + 1) * NKY + ncol];
    bi.x = Bi[(size_t)kk * NKY + ncol];
    bi.y = Bi[(size_t)(kk + 1) * NKY + ncol];
    aR = wmma4(ar, br, aR);
    aR = wmma4(ain, bi, aR);
    aI = wmma4(ar, bi, aI);
    aI = wmma4(ai, br, aI);
  }
  for (int r = 0; r < 8; ++r) {
    int xx = mt * 16 + r + 8 * half;
    size_t o = ((size_t)bo * NS + xx) * NKY + ncol;
    Zre[o] = aR[r];
    Zim[o] = aI[r];
  }
}

// ---------------------------------------------------------------------------
// Stage G: inverse row irfft (Re-projection, weights+1/N in table) fused with
// GN2 stats. One wave = 16 rows; loops 16 yy-tiles, A (Z) cached in VGPRs.
// ---------------------------------------------------------------------------
__global__ void k_irowdft(const float* __restrict__ Zre,
                          const float* __restrict__ Zim,
                          const float* __restrict__ Gr,
                          const float* __restrict__ Gi, float* __restrict__ out,
                          double* gsum2) {
  int wave = blockIdx.x * 8 + (threadIdx.x >> 5);  // 16384 tiles
  int lane = threadIdx.x & 31, l16 = lane & 15, half = lane >> 4;
  int row0 = wave * 16;
  int myrow = row0 + l16;
  const float* zr = Zre + (size_t)myrow * NKY;
  const float* zi = Zim + (size_t)myrow * NKY;
  v2f ar[8], ai[8];
  for (int s = 0; s < 8; ++s) {
    int kk = 4 * s + 2 * half;
    ar[s] = *reinterpret_cast<const v2f*>(zr + kk);
    ai[s] = *reinterpret_cast<const v2f*>(zi + kk);
  }
  double lsum = 0.0, lsum2 = 0.0;
  for (int ntile = 0; ntile < 16; ++ntile) {
    int ncol = ntile * 16 + l16;
    v8f acc = {};
    for (int s = 0; s < 8; ++s) {
      int kk = 4 * s + 2 * half;
      v2f br, bi;
      br.x = Gr[kk * NS + ncol];
      br.y = Gr[(kk + 1) * NS + ncol];
      bi.x = Gi[kk * NS + ncol];
      bi.y = Gi[(kk + 1) * NS + ncol];
      acc = wmma4(ar[s], br, acc);   // Zr * (c*cos/N)
      acc = wmma4(ai[s], bi, acc);   // Zi * (-c*sin/N)
    }
    for (int r = 0; r < 8; ++r) {
      int rr = row0 + r + 8 * half;
      float v = acc[r];
      out[(size_t)rr * NS + ncol] = v;
      lsum += (double)v;
      lsum2 += (double)v * (double)v;
    }
  }
  int b = blockIdx.x >> 7;  // 128 blocks (=16384 rows) per batch
  block_reduce_atomic(lsum, lsum2, &gsum2[b * 2], &gsum2[b * 2 + 1]);
}

// Stage H: GN2 apply in place on d_out.
__global__ void k_gn2(float* __restrict__ out, const float* __restrict__ stats,
                      const float* __restrict__ gw, const float* __restrict__ gb) {
  size_t i = ((size_t)blockIdx.x * 256 + threadIdx.x) * 4;  // 67108864 total
  int b = (int)(i >> 22);
  int c = (int)((i >> 16) & 63);
  float mu = stats[b * 2], istd = stats[b * 2 + 1];
  float sc = istd * gw[c];
  float bs = gb[c] - mu * sc;
  float4 v = *reinterpret_cast<float4*>(out + i);
  v.x = v.x * sc + bs;
  v.y = v.y * sc + bs;
  v.z = v.z * sc + bs;
  v.w = v.w * sc + bs;
  *reinterpret_cast<float4*>(out + i) = v;
}

// ---------------------------------------------------------------------------
extern "C" void kernel_launch(void* const* d_in, const int* in_sizes, int n_in,
                              void* d_out, int out_size, void* d_ws,
                              size_t ws_size, hipStream_t stream) {
  (void)in_sizes; (void)n_in; (void)out_size; (void)ws_size;
  const float* x = (const float*)d_in[0];
  const float* ReN = (const float*)d_in[1];
  const float* gn1_w = (const float*)d_in[2];
  const float* gn1_b = (const float*)d_in[3];
  const float* p_w1 = (const float*)d_in[4];
  const float* p_b1 = (const float*)d_in[5];
  const float* p_w2 = (const float*)d_in[6];
  const float* p_b2 = (const float*)d_in[7];
  const float2* w3 = (const float2*)d_in[8];
  const float2* w4 = (const float2*)d_in[9];
  const float* gn2_w = (const float*)d_in[10];
  const float* gn2_b = (const float*)d_in[11];
  float* out = (float*)d_out;
  float* ws = (float*)d_ws;
  double* gsum1 = (double*)d_ws;       // 32 doubles
  double* gsum2 = gsum1 + 32;          // 32 doubles

  hipMemsetAsync(d_ws, 0, 512, stream);  // zero both stat accumulators

  k_tables<<<64, 256, 0, stream>>>(ws);

  k_stats1<<<dim3(256, 16), 256, 0, stream>>>(x, gsum1);
  k_finalize<<<1, 16, 0, stream>>>(gsum1, ws + OFF_STATS1, 1.0 / 4194304.0);

  k_rowdft<<<2048, 256, 0, stream>>>(x, ws + OFF_STATS1, gn1_w, gn1_b,
                                     ws + OFF_TBROW_RE, ws + OFF_TBROW_IM,
                                     ws + OFF_Y1RE, ws + OFF_Y1IM);

  k_coldft<<<1024, 256, 0, stream>>>(ws + OFF_Y1RE, ws + OFF_Y1IM,
                                     ws + OFF_TCCOL_RE, ws + OFF_TCCOL_IM,
                                     ws + OFF_Y2RE, ws + OFF_Y2IM);

  k_feature<<<128, 256, 0, stream>>>(ReN, p_w1, p_b1, p_w2, p_b2,
                                     ws + OFF_FEAT);
  k_mulfeat<<<8192, 256, 0, stream>>>(ws);

  k_specmul<<<1024, 256, 0, stream>>>(ws + OFF_Y2RE, ws + OFF_Y2IM, w3, w4,
                                      ws + OFF_ORE, ws + OFF_OIM);

  k_icoldft<<<4096, 256, 0, stream>>>(ws + OFF_ORE, ws + OFF_OIM,
                                      ws + OFF_TFCOL_RE, ws + OFF_TFCOL_IM,
                                      ws + OFF_ZRE, ws + OFF_ZIM);

  k_irowdft<<<2048, 256, 0, stream>>>(ws + OFF_ZRE, ws + OFF_ZIM,
                                      ws + OFF_TGROW_RE, ws + OFF_TGROW_IM, out,
                                      gsum2);
  k_finalize<<<1, 16, 0, stream>>>(gsum2, ws + OFF_STATS2, 1.0 / 4194304.0);

  k_gn2<<<65536, 256, 0, stream>>>(out, ws + OFF_STATS2, gn2_w, gn2_b);
}